// LineFlowLayer_49675591745745
// MI455X (gfx1250) — compile-verified
//
#include <hip/hip_runtime.h>
#include <math.h>

// Problem constants (match reference)
#define NB      20000            // num_buses
#define OTHER   1000
#define ROW     (2 * NB + OTHER) // 41000 floats per x row
#define NLINES  320000
#define BATCH   128
#define NSPLIT  4                // line segments per batch row
#define SEG     (NLINES / NSPLIT)   // 80000, divisible by 4
#define OUT0    (BATCH * ROW)       // floats in `out`; flows2 follows

typedef float f32x4 __attribute__((ext_vector_type(4)));
typedef int   i32x4 __attribute__((ext_vector_type(4)));

// ---------------------------------------------------------------------------
// CDNA5 async DMA helpers (ASYNCcnt-tracked, gfx1250).
// Generic-pointer low 32 bits == wave-relative LDS byte address (LDS aperture).
// GVS addressing: SGPR64 base + per-lane VGPR32 offset, 16B per lane per issue.
// ---------------------------------------------------------------------------
__device__ __forceinline__ void lds_stage_issue(float* s_dst, const float* g_src,
                                                int nbytes) {
  unsigned lds_base = (unsigned)(unsigned long long)s_dst;
  unsigned long long gbase = (unsigned long long)g_src;
  for (int off = (int)threadIdx.x * 16; off < nbytes; off += (int)blockDim.x * 16) {
    unsigned lds_addr = lds_base + (unsigned)off;
    asm volatile("global_load_async_to_lds_b128 %0, %1, %2"
                 :: "v"(lds_addr), "v"(off), "s"(gbase)
                 : "memory");
  }
}

__device__ __forceinline__ void lds_store_issue(float* g_dst, const float* s_src,
                                                int nbytes) {
  unsigned lds_base = (unsigned)(unsigned long long)s_src;
  unsigned long long gbase = (unsigned long long)g_dst;
  for (int off = (int)threadIdx.x * 16; off < nbytes; off += (int)blockDim.x * 16) {
    unsigned lds_addr = lds_base + (unsigned)off;
    asm volatile("global_store_async_from_lds_b128 %0, %1, %2"
                 :: "v"(off), "v"(lds_addr), "s"(gbase)
                 : "memory");
  }
}

__device__ __forceinline__ void wait_async0() {
#if __has_builtin(__builtin_amdgcn_s_wait_asynccnt)
  __builtin_amdgcn_s_wait_asynccnt(0);
#else
  asm volatile("s_wait_asynccnt 0" ::: "memory");
#endif
}

// ---------------------------------------------------------------------------
// K1: copy voltages [0,NB) and other-params [2NB,ROW) of each row to `out`
// (the angle region [NB,2NB) is produced later by the fused flows kernel);
// zero the delta scratch.
// ---------------------------------------------------------------------------
__global__ void lf_init_kernel(const float* __restrict__ x,
                               float* __restrict__ out,
                               float* __restrict__ delta) {
  const int stride = (int)(gridDim.x * blockDim.x);
  const int i0 = (int)(blockIdx.x * blockDim.x + threadIdx.x);

  const int NV1 = BATCH * (NB / 4);       // voltages: 640,000 vec4
  for (int v = i0; v < NV1; v += stride) {
    const int b = v / (NB / 4);
    const int j = (v % (NB / 4)) * 4;
    const size_t o = (size_t)b * ROW + j;
    f32x4 a = __builtin_nontemporal_load((const f32x4*)(x + o));
    __builtin_nontemporal_store(a, (f32x4*)(out + o));
  }
  const int NV2 = BATCH * (OTHER / 4);    // other params: 32,000 vec4
  for (int v = i0; v < NV2; v += stride) {
    const int b = v / (OTHER / 4);
    const int j = (v % (OTHER / 4)) * 4;
    const size_t o = (size_t)b * ROW + 2 * NB + j;
    f32x4 a = __builtin_nontemporal_load((const f32x4*)(x + o));
    __builtin_nontemporal_store(a, (f32x4*)(out + o));
  }
  const int NZV = (BATCH * NB) / 4;       // delta: 640,000 vec4
  f32x4 z = {0.f, 0.f, 0.f, 0.f};
  for (int v = i0; v < NZV; v += stride)
    ((f32x4*)delta)[v] = z;
}

// ---------------------------------------------------------------------------
// K2: per (row, segment): async-stage the angle row to LDS, accumulate clamp
// adjustments into an LDS delta table (ds_add_f32), flush with global fp32
// atomics (8x fewer L2 atomics than naive scatter).
// ---------------------------------------------------------------------------
__global__ void __launch_bounds__(1024)
lf_accum_kernel(const float* __restrict__ x,
                const int* __restrict__ fidx,
                const int* __restrict__ tidx,
                const float* __restrict__ react,
                const float* __restrict__ limit,
                float* __restrict__ delta) {
  __shared__ float s_ang[NB];     // 80 KB
  __shared__ float s_delta[NB];   // 80 KB
  const int row = (int)blockIdx.x / NSPLIT;
  const int seg = (int)blockIdx.x % NSPLIT;

  lds_stage_issue(s_ang, x + (size_t)row * ROW + NB, NB * 4);
  for (int i = (int)threadIdx.x; i < NB; i += (int)blockDim.x)
    s_delta[i] = 0.0f;
  wait_async0();
  __syncthreads();

  const int vbeg = seg * (SEG / 4);
  const int vend = vbeg + (SEG / 4);
  for (int v = vbeg + (int)threadIdx.x; v < vend; v += (int)blockDim.x) {
    __builtin_prefetch((const i32x4*)fidx + v + (int)blockDim.x, 0, 1);
    i32x4 f  = __builtin_nontemporal_load((const i32x4*)fidx  + v);
    i32x4 t  = __builtin_nontemporal_load((const i32x4*)tidx  + v);
    f32x4 r  = __builtin_nontemporal_load((const f32x4*)react + v);
    f32x4 lm = __builtin_nontemporal_load((const f32x4*)limit + v);
#pragma unroll
    for (int k = 0; k < 4; ++k) {
      float ad = s_ang[f[k]] - s_ang[t[k]];
      float rl = r[k] * lm[k];
      float half = (fabsf(ad) > rl) ? 0.5f * (copysignf(rl, ad) - ad) : 0.0f;
      atomicAdd(&s_delta[f[k]], half);   // ds_add_f32
      atomicAdd(&s_delta[t[k]], half);   // ds_add_f32
    }
  }
  __syncthreads();

  float* drow = delta + (size_t)row * NB;
  for (int i = (int)threadIdx.x; i < NB; i += (int)blockDim.x)
    atomicAdd(&drow[i], s_delta[i]);     // global_atomic_add_f32
}

// ---------------------------------------------------------------------------
// K3 (fused): per (row, segment):
//   1. async-stage x angles AND delta row into LDS (two DMA streams)
//   2. angles2 row in LDS: s_ang += s_delta
//   3. seg==0 block streams angles2 back to `out` via async LDS->global DMA
//   4. all blocks: LDS gathers -> flows2, NT B128 streaming stores
// ---------------------------------------------------------------------------
__global__ void __launch_bounds__(1024)
lf_flows_kernel(const float* __restrict__ x,
                const float* __restrict__ delta,
                const int* __restrict__ fidx,
                const int* __restrict__ tidx,
                const float* __restrict__ react,
                float* __restrict__ out,
                float* __restrict__ flows2) {
  __shared__ float s_ang[NB];     // 80 KB -> becomes angles2
  __shared__ float s_d[NB];       // 80 KB
  const int row = (int)blockIdx.x / NSPLIT;
  const int seg = (int)blockIdx.x % NSPLIT;

  lds_stage_issue(s_ang, x + (size_t)row * ROW + NB, NB * 4);
  lds_stage_issue(s_d, delta + (size_t)row * NB, NB * 4);
  wait_async0();
  __syncthreads();

  for (int i = (int)threadIdx.x; i < NB; i += (int)blockDim.x)
    s_ang[i] += s_d[i];
  __syncthreads();

  if (seg == 0)  // one block per row publishes angles2 (LDS -> global DMA)
    lds_store_issue(out + (size_t)row * ROW + NB, s_ang, NB * 4);

  float* frow = flows2 + (size_t)row * NLINES;
  const int vbeg = seg * (SEG / 4);
  const int vend = vbeg + (SEG / 4);
  for (int v = vbeg + (int)threadIdx.x; v < vend; v += (int)blockDim.x) {
    __builtin_prefetch((const i32x4*)fidx + v + (int)blockDim.x, 0, 1);
    i32x4 f = __builtin_nontemporal_load((const i32x4*)fidx  + v);
    i32x4 t = __builtin_nontemporal_load((const i32x4*)tidx  + v);
    f32x4 r = __builtin_nontemporal_load((const f32x4*)react + v);
    f32x4 res;
#pragma unroll
    for (int k = 0; k < 4; ++k)
      res[k] = (s_ang[f[k]] - s_ang[t[k]]) / r[k];
    __builtin_nontemporal_store(res, (f32x4*)frow + v);
  }
  wait_async0();  // ensure LDS->global angles2 DMA retired before wave exit
}

// ---------------------------------------------------------------------------
extern "C" void kernel_launch(void* const* d_in, const int* in_sizes, int n_in,
                              void* d_out, int out_size, void* d_ws, size_t ws_size,
                              hipStream_t stream) {
  const float* x     = (const float*)d_in[0];
  const int*   fidx  = (const int*)d_in[1];
  const int*   tidx  = (const int*)d_in[2];
  const float* react = (const float*)d_in[3];
  const float* limit = (const float*)d_in[4];
  // d_in[5] = num_buses scalar (compile-time constant NB)

  float* out    = (float*)d_out;
  float* flows2 = out + OUT0;            // second tuple output, concatenated
  float* delta  = (float*)d_ws;          // BATCH*NB floats = 10.24 MB scratch

  lf_init_kernel <<<2048, 256, 0, stream>>>(x, out, delta);
  lf_accum_kernel<<<BATCH * NSPLIT, 1024, 0, stream>>>(x, fidx, tidx, react, limit, delta);
  lf_flows_kernel<<<BATCH * NSPLIT, 1024, 0, stream>>>(x, delta, fidx, tidx, react, out, flows2);
}